// GraphTransformerLayer_42288247996466
// MI455X (gfx1250) — compile-verified
//
#include <hip/hip_runtime.h>
#include <hip/hip_bf16.h>

#define D_MODEL   128
#define N_HEADS   8
#define HEAD_DIM  16

typedef float v2f __attribute__((ext_vector_type(2)));
typedef float v8f __attribute__((ext_vector_type(8)));

// ---------------------------------------------------------------------------
// Phase 1: per-node Q/K/V projection, Y = X @ W^T + b, via V_WMMA_F32_16X16X4_F32
// One wave computes one 16x16 output tile; waves 0..7 of a block cover the 8
// column tiles of D_MODEL=128; blockIdx.x covers node tiles (50000/16 = 3125).
// A-tile layout (ISA 7.12.2, 32-bit A 16x4): lane<16 -> K={k,k+1}, lane>=16 ->
// K={k+2,k+3}, M = lane&15.  B-tile mirrored: N = lane&15, K split the same way.
// ---------------------------------------------------------------------------
__global__ __launch_bounds__(256) void qkv_gemm_wmma(
    const float* __restrict__ X,
    const float* __restrict__ Wq, const float* __restrict__ bq,
    const float* __restrict__ Wk, const float* __restrict__ bk,
    const float* __restrict__ Wv, const float* __restrict__ bv,
    float* __restrict__ Qo, float* __restrict__ Ko, float* __restrict__ Vo)
{
    const int lane  = threadIdx.x & 31;
    const int wave  = threadIdx.x >> 5;
    const int m0    = blockIdx.x << 4;        // node tile base
    const int j0    = wave << 4;              // output-column tile base
    const int mrow  = m0 + (lane & 15);       // A row this lane feeds
    const int jcol  = j0 + (lane & 15);       // B column this lane feeds
    const int khalf = (lane >> 4) << 1;       // 0 for lanes 0-15, 2 for 16-31

    v8f accq = {}; v8f acck = {}; v8f accv = {};

    const float* __restrict__ xrow  = X  + (size_t)mrow * D_MODEL;
    const float* __restrict__ wqrow = Wq + (size_t)jcol * D_MODEL;
    const float* __restrict__ wkrow = Wk + (size_t)jcol * D_MODEL;
    const float* __restrict__ wvrow = Wv + (size_t)jcol * D_MODEL;

    #pragma unroll 4
    for (int k = 0; k < D_MODEL; k += 4) {
        const int ka = k + khalf;
        v2f a   = *(const v2f*)(xrow  + ka);
        v2f bq2 = *(const v2f*)(wqrow + ka);
        v2f bk2 = *(const v2f*)(wkrow + ka);
        v2f bv2 = *(const v2f*)(wvrow + ka);
        accq = __builtin_amdgcn_wmma_f32_16x16x4_f32(false, a, false, bq2,
                                                     (short)0, accq, false, false);
        acck = __builtin_amdgcn_wmma_f32_16x16x4_f32(false, a, false, bk2,
                                                     (short)0, acck, false, false);
        accv = __builtin_amdgcn_wmma_f32_16x16x4_f32(false, a, false, bv2,
                                                     (short)0, accv, false, false);
    }

    const float biasq = bq[jcol];
    const float biask = bk[jcol];
    const float biasv = bv[jcol];

    // C/D layout: lanes 0-15 -> M = i, lanes 16-31 -> M = 8+i; N = lane&15.
    const int rbase = m0 + ((lane >> 4) << 3);
    #pragma unroll
    for (int i = 0; i < 8; ++i) {
        const size_t off = (size_t)(rbase + i) * D_MODEL + jcol;
        Qo[off] = accq[i] + biasq;
        Ko[off] = acck[i] + biask;
        Vo[off] = accv[i] + biasv;
    }
}

// ---------------------------------------------------------------------------
// Phase 2a: per-(edge, head) attention logits -> exp -> per-(node, head) sums.
// 8 consecutive lanes share one edge (heads 0..7), so the 512B q/k rows are
// read with coalesced 128b loads that hit L2 (QKV fit in the 192MB L2).
// ---------------------------------------------------------------------------
__global__ __launch_bounds__(256) void edge_attention(
    const int* __restrict__ eidx,          // [2, E] (rows then cols)
    const float* __restrict__ Q, const float* __restrict__ K,
    float* __restrict__ expAtt,            // [E, 8]
    float* __restrict__ attSum,            // [N, 8]
    int nEdges)
{
    const int t = blockIdx.x * blockDim.x + threadIdx.x;
    if (t >= nEdges * N_HEADS) return;
    const int e = t >> 3;
    const int h = t & 7;
    const int row = eidx[e];
    const int col = eidx[nEdges + e];

    const float4* __restrict__ q =
        (const float4*)(Q + (size_t)row * D_MODEL + h * HEAD_DIM);
    const float4* __restrict__ k =
        (const float4*)(K + (size_t)col * D_MODEL + h * HEAD_DIM);

    float dot = 0.0f;
    #pragma unroll
    for (int i = 0; i < 4; ++i) {
        float4 a = q[i], b = k[i];
        dot = fmaf(a.x, b.x, dot);
        dot = fmaf(a.y, b.y, dot);
        dot = fmaf(a.z, b.z, dot);
        dot = fmaf(a.w, b.w, dot);
    }
    float att = dot * 0.25f;                       // 1/sqrt(HEAD_DIM)
    att = fminf(10.0f, fmaxf(-10.0f, att));
    const float ea = expf(att);
    expAtt[t] = ea;
    unsafeAtomicAdd(attSum + (size_t)row * N_HEADS + h, ea);  // global_atomic_add_f32
}

// ---------------------------------------------------------------------------
// Phase 2b: normalize and scatter norm * v into the accumulator (d_out).
// ---------------------------------------------------------------------------
__global__ __launch_bounds__(256) void edge_aggregate(
    const int* __restrict__ eidx,
    const float* __restrict__ V,
    const float* __restrict__ expAtt,
    const float* __restrict__ attSum,
    float* __restrict__ acc,               // [N, 128], pre-zeroed
    int nEdges)
{
    const int t = blockIdx.x * blockDim.x + threadIdx.x;
    if (t >= nEdges * N_HEADS) return;
    const int e = t >> 3;
    const int h = t & 7;
    const int row = eidx[e];
    const int col = eidx[nEdges + e];

    const float norm = expAtt[t] /
                       (attSum[(size_t)row * N_HEADS + h] + 1e-8f);

    const float4* __restrict__ v =
        (const float4*)(V + (size_t)col * D_MODEL + h * HEAD_DIM);
    float* __restrict__ dst = acc + (size_t)row * D_MODEL + h * HEAD_DIM;

    #pragma unroll
    for (int i = 0; i < 4; ++i) {
        float4 vv = v[i];
        unsafeAtomicAdd(dst + 4 * i + 0, norm * vv.x);
        unsafeAtomicAdd(dst + 4 * i + 1, norm * vv.y);
        unsafeAtomicAdd(dst + 4 * i + 2, norm * vv.z);
        unsafeAtomicAdd(dst + 4 * i + 3, norm * vv.w);
    }
}

// ---------------------------------------------------------------------------
// Phase 3: residual add + LayerNorm (eps 1e-6), in place on d_out.
// One wave (32 lanes) per node, 4 channels per lane, shfl_xor reductions.
// ---------------------------------------------------------------------------
__global__ __launch_bounds__(256) void residual_layernorm(
    float* __restrict__ out,               // [N, 128]: in = segment_sum, out = LN
    const float* __restrict__ X,
    const float* __restrict__ gamma, const float* __restrict__ beta,
    int nNodes)
{
    const int wave = threadIdx.x >> 5;
    const int lane = threadIdx.x & 31;
    const int node = blockIdx.x * 8 + wave;
    if (node >= nNodes) return;

    const size_t base = (size_t)node * D_MODEL + lane * 4;
    float4 v = *(const float4*)(out + base);
    float4 x = *(const float4*)(X + base);
    v.x += x.x; v.y += x.y; v.z += x.z; v.w += x.w;

    float s = v.x + v.y + v.z + v.w;
    #pragma unroll
    for (int off = 16; off > 0; off >>= 1) s += __shfl_xor(s, off, 32);
    const float mu = s * (1.0f / D_MODEL);

    const float dx = v.x - mu, dy = v.y - mu, dz = v.z - mu, dw = v.w - mu;
    float sv = dx * dx + dy * dy + dz * dz + dw * dw;
    #pragma unroll
    for (int off = 16; off > 0; off >>= 1) sv += __shfl_xor(sv, off, 32);
    const float inv = rsqrtf(sv * (1.0f / D_MODEL) + 1e-6f);

    float4 g = *(const float4*)(gamma + lane * 4);
    float4 b = *(const float4*)(beta + lane * 4);
    float4 r;
    r.x = dx * inv * g.x + b.x;
    r.y = dy * inv * g.y + b.y;
    r.z = dz * inv * g.z + b.z;
    r.w = dw * inv * g.w + b.w;
    *(float4*)(out + base) = r;
}

extern "C" void kernel_launch(void* const* d_in, const int* in_sizes, int n_in,
                              void* d_out, int out_size, void* d_ws, size_t ws_size,
                              hipStream_t stream)
{
    const float* X     = (const float*)d_in[0];   // node_embeds [N,128]
    const int*   eidx  = (const int*)  d_in[1];   // edge_index  [2,E]
    const float* Wq    = (const float*)d_in[2];
    const float* bq    = (const float*)d_in[3];
    const float* Wk    = (const float*)d_in[4];
    const float* bk    = (const float*)d_in[5];
    const float* Wv    = (const float*)d_in[6];
    const float* bv    = (const float*)d_in[7];
    const float* gamma = (const float*)d_in[8];
    const float* beta  = (const float*)d_in[9];
    float* out = (float*)d_out;

    const int N = in_sizes[0] / D_MODEL;          // 50000
    const int E = in_sizes[1] / 2;                // 640000

    // Workspace layout (floats): Q | K | V | attSum | expAtt  (~99 MB)
    float* Q      = (float*)d_ws;
    float* K      = Q + (size_t)N * D_MODEL;
    float* V      = K + (size_t)N * D_MODEL;
    float* attSum = V + (size_t)N * D_MODEL;
    float* expAtt = attSum + (size_t)N * N_HEADS;

    hipMemsetAsync(attSum, 0, (size_t)N * N_HEADS * sizeof(float), stream);
    hipMemsetAsync(out, 0, (size_t)N * D_MODEL * sizeof(float), stream);

    // Phase 1: WMMA QKV projection. 8 waves/block = 8 column tiles.
    qkv_gemm_wmma<<<(N + 15) / 16, 256, 0, stream>>>(
        X, Wq, bq, Wk, bk, Wv, bv, Q, K, V);

    // Phase 2a: logits + softmax denominators.
    const int edgeThreads = E * N_HEADS;
    edge_attention<<<(edgeThreads + 255) / 256, 256, 0, stream>>>(
        eidx, Q, K, expAtt, attSum, E);

    // Phase 2b: normalized weighted-V scatter.
    edge_aggregate<<<(edgeThreads + 255) / 256, 256, 0, stream>>>(
        eidx, V, expAtt, attSum, out, E);

    // Phase 3: residual + LayerNorm.
    residual_layernorm<<<(N + 7) / 8, 256, 0, stream>>>(
        out, X, gamma, beta, N);
}